// EntmaxAttention_15616501088533
// MI455X (gfx1250) — compile-verified
//
#include <hip/hip_runtime.h>

typedef __attribute__((ext_vector_type(2))) float v2f;
typedef __attribute__((ext_vector_type(8))) float v8f;

#define SEQ    2048
#define HD     64
#define NH     16
#define NITER  50

// LDS layout (floats)
#define LDS_PSTRIDE  2050                       // 2048 + 2 pad (bank-conflict avoidance)
#define LDS_P        0
#define LDS_REDA     (16 * LDS_PSTRIDE)         // [8 waves][16 rows]
#define LDS_REDB     (LDS_REDA + 128)           // [8 waves][16 rows]
#define LDS_ROWA     (LDS_REDB + 128)           // [16]
#define LDS_ROWB     (LDS_ROWA + 16)            // [16]
#define LDS_OUTBUF   (LDS_ROWB + 16)            // [4 tiles][16][16]
#define LDS_FLOATS   (LDS_OUTBUF + 1024)
#define LDS_BYTES    (LDS_FLOATS * 4)

__device__ __forceinline__ v8f wmma_f32(v2f a, v2f b, v8f c) {
    return __builtin_amdgcn_wmma_f32_16x16x4_f32(false, a, false, b, (short)0, c,
                                                 false, false);
}

extern "C" __global__ __launch_bounds__(256, 1)
void entmax_attn_kernel(const float* __restrict__ Q, const float* __restrict__ K,
                        const float* __restrict__ V, float* __restrict__ Out,
                        float* __restrict__ StuA, float* __restrict__ TeaA)
{
    extern __shared__ float smem[];
    float* Plds   = smem + LDS_P;
    float* redA   = smem + LDS_REDA;
    float* redB   = smem + LDS_REDB;
    float* rowA   = smem + LDS_ROWA;
    float* rowB   = smem + LDS_ROWB;
    float* outbuf = smem + LDS_OUTBUF;

    const int tid   = threadIdx.x;
    const int lane  = tid & 31;            // wave32
    const int wave  = tid >> 5;            // 8 waves
    const int lq    = lane & 15;           // column-in-tile / A row
    const int half  = lane >> 4;
    const int rbase = half * 8;            // C/D fragment rows rbase..rbase+7
    const int ko    = half * 2;            // A/B k-pair offset within k4

    const int h  = blockIdx.x >> 7;        // 128 q-tiles per head
    const int q0 = (blockIdx.x & 127) * 16;

    const float scale = 0.125f;            // 1/sqrt(64)

    const float* Qh = Q + (size_t)h * SEQ * HD;
    const float* Kh = K + (size_t)h * SEQ * HD;
    const float* Vh = V + (size_t)h * SEQ * HD;

    // ---------------- A fragments: Q tile 16x64 (16 k-chunks of 4) ----------
    v2f a[16];
    {
        const float* qrow = Qh + (size_t)(q0 + lq) * HD + ko;
        #pragma unroll
        for (int kk = 0; kk < 16; ++kk)
            a[kk] = *(const v2f*)(qrow + 4 * kk);
    }

    // ---------------- scores: this wave's 16 groups of 16 keys --------------
    const int kw0 = wave * 256;
    v8f acc[16];
    #pragma unroll
    for (int g = 0; g < 16; ++g) {
        const float* krow = Kh + (size_t)(kw0 + g * 16 + lq) * HD + ko;
        v2f b[16];
        #pragma unroll
        for (int kk = 0; kk < 16; ++kk)
            b[kk] = *(const v2f*)(krow + 4 * kk);
        v8f c = {0.f, 0.f, 0.f, 0.f, 0.f, 0.f, 0.f, 0.f};
        #pragma unroll
        for (int kk = 0; kk < 16; ++kk)
            c = wmma_f32(a[kk], b[kk], c);     // v_wmma_f32_16x16x4_f32 chain
        acc[g] = c;
    }

    // ---------------- row max / min (raw dot products) ----------------------
    float rm[8], rn[8];
    #pragma unroll
    for (int r = 0; r < 8; ++r) { rm[r] = -3.4e38f; rn[r] = 3.4e38f; }
    #pragma unroll
    for (int g = 0; g < 16; ++g)
        #pragma unroll
        for (int r = 0; r < 8; ++r) {
            rm[r] = fmaxf(rm[r], acc[g][r]);
            rn[r] = fminf(rn[r], acc[g][r]);
        }
    #pragma unroll
    for (int m = 1; m <= 8; m <<= 1)
        #pragma unroll
        for (int r = 0; r < 8; ++r) {
            rm[r] = fmaxf(rm[r], __shfl_xor(rm[r], m, 32));
            rn[r] = fminf(rn[r], __shfl_xor(rn[r], m, 32));
        }
    if (lq == 0)
        #pragma unroll
        for (int r = 0; r < 8; ++r) {
            redA[wave * 16 + rbase + r] = rm[r];
            redB[wave * 16 + rbase + r] = rn[r];
        }
    __syncthreads();
    if (tid < 16) {
        float mx = redA[tid], mn = redB[tid];
        for (int w = 1; w < 8; ++w) {
            mx = fmaxf(mx, redA[w * 16 + tid]);
            mn = fminf(mn, redB[w * 16 + tid]);
        }
        rowA[tid] = mx;
        rowB[tid] = mn;
    }
    __syncthreads();

    // xa = 0.5 * (dot - rowmax) * scale ; tau range per reference
    float tmin[8], tmax[8];
    #pragma unroll
    for (int r = 0; r < 8; ++r) {
        float RM = rowA[rbase + r];
        float RN = rowB[rbase + r];
        tmin[r] = (RN - RM) * scale - 1.0f;
        tmax[r] = 0.0f;
        rm[r]   = RM;
    }
    #pragma unroll
    for (int g = 0; g < 16; ++g)
        #pragma unroll
        for (int r = 0; r < 8; ++r)
            acc[g][r] = (acc[g][r] - rm[r]) * (0.5f * scale);

    // ---------------- bisection (50 iters, scores stay in VGPRs) ------------
    float tau[8];
    #pragma unroll 1
    for (int it = 0; it < NITER; ++it) {
        float s[8];
        #pragma unroll
        for (int r = 0; r < 8; ++r) {
            tau[r] = 0.5f * (tmin[r] + tmax[r]);
            s[r]   = 0.0f;
        }
        #pragma unroll
        for (int g = 0; g < 16; ++g)
            #pragma unroll
            for (int r = 0; r < 8; ++r) {
                float d = fmaxf(acc[g][r] - tau[r], 0.0f);
                s[r] = fmaf(d, d, s[r]);       // expo = 1/(alpha-1) = 2
            }
        #pragma unroll
        for (int m = 1; m <= 8; m <<= 1)
            #pragma unroll
            for (int r = 0; r < 8; ++r)
                s[r] += __shfl_xor(s[r], m, 32);
        if (lq == 0)
            #pragma unroll
            for (int r = 0; r < 8; ++r)
                redA[wave * 16 + rbase + r] = s[r];
        __syncthreads();
        if (tid < 16) {
            float t = redA[tid];
            for (int w = 1; w < 8; ++w) t += redA[w * 16 + tid];
            rowA[tid] = t;
        }
        __syncthreads();
        #pragma unroll
        for (int r = 0; r < 8; ++r) {
            float Sr = rowA[rbase + r];
            tmin[r] = (Sr > 1.0f) ? tau[r] : tmin[r];
            tmax[r] = (Sr < 1.0f) ? tau[r] : tmax[r];
        }
    }
    #pragma unroll
    for (int r = 0; r < 8; ++r) tau[r] = 0.5f * (tmin[r] + tmax[r]);

    // ---------------- teacher softmax (x = 2*xa) -----------------------------
    {
        float es[8];
        #pragma unroll
        for (int r = 0; r < 8; ++r) es[r] = 0.0f;
        #pragma unroll
        for (int g = 0; g < 16; ++g)
            #pragma unroll
            for (int r = 0; r < 8; ++r)
                es[r] += __expf(2.0f * acc[g][r]);
        #pragma unroll
        for (int m = 1; m <= 8; m <<= 1)
            #pragma unroll
            for (int r = 0; r < 8; ++r)
                es[r] += __shfl_xor(es[r], m, 32);
        if (lq == 0)
            #pragma unroll
            for (int r = 0; r < 8; ++r)
                redA[wave * 16 + rbase + r] = es[r];
        __syncthreads();
        if (tid < 16) {
            float t = redA[tid];
            for (int w = 1; w < 8; ++w) t += redA[w * 16 + tid];
            rowA[tid] = t;
        }
        __syncthreads();
        float inv[8];
        #pragma unroll
        for (int r = 0; r < 8; ++r) inv[r] = 1.0f / rowA[rbase + r];
        float* tptr = TeaA + ((size_t)(h * SEQ + q0 + rbase)) * SEQ + kw0 + lq;
        #pragma unroll
        for (int g = 0; g < 16; ++g)
            #pragma unroll
            for (int r = 0; r < 8; ++r) {
                float e = __expf(2.0f * acc[g][r]) * inv[r];
                __builtin_nontemporal_store(e, tptr + (size_t)r * SEQ + g * 16);
            }
    }

    // ---------------- student entmax probabilities ---------------------------
    {
        float ps[8];
        #pragma unroll
        for (int r = 0; r < 8; ++r) ps[r] = 0.0f;
        #pragma unroll
        for (int g = 0; g < 16; ++g)
            #pragma unroll
            for (int r = 0; r < 8; ++r) {
                float d = fmaxf(acc[g][r] - tau[r], 0.0f);
                d = d * d;
                acc[g][r] = d;
                ps[r] += d;
            }
        #pragma unroll
        for (int m = 1; m <= 8; m <<= 1)
            #pragma unroll
            for (int r = 0; r < 8; ++r)
                ps[r] += __shfl_xor(ps[r], m, 32);
        if (lq == 0)
            #pragma unroll
            for (int r = 0; r < 8; ++r)
                redA[wave * 16 + rbase + r] = ps[r];
        __syncthreads();
        if (tid < 16) {
            float t = redA[tid];
            for (int w = 1; w < 8; ++w) t += redA[w * 16 + tid];
            rowA[tid] = t;
        }
        __syncthreads();
        float inv[8];
        #pragma unroll
        for (int r = 0; r < 8; ++r) inv[r] = 1.0f / rowA[rbase + r];
        float* sptr = StuA + ((size_t)(h * SEQ + q0 + rbase)) * SEQ + kw0 + lq;
        #pragma unroll
        for (int g = 0; g < 16; ++g)
            #pragma unroll
            for (int r = 0; r < 8; ++r) {
                float p = acc[g][r] * inv[r];
                __builtin_nontemporal_store(p, sptr + (size_t)r * SEQ + g * 16);
                Plds[(rbase + r) * LDS_PSTRIDE + kw0 + g * 16 + lq] = p;
            }
    }
    __syncthreads();

    // ---------------- output = P @ V via WMMA --------------------------------
    // wave w: output-col tile n0=(w&3)*16, key half kb=(w>>2)*1024
    {
        const int n0 = (wave & 3) * 16;
        const int kb = (wave >> 2) * 1024;
        v8f o = {0.f, 0.f, 0.f, 0.f, 0.f, 0.f, 0.f, 0.f};
        #pragma unroll 4
        for (int k4 = 0; k4 < 256; ++k4) {
            const int k = kb + k4 * 4;
            v2f pa = *(const v2f*)(&Plds[lq * LDS_PSTRIDE + k + ko]);
            v2f pb;
            pb.x = Vh[(size_t)(k + ko) * HD + n0 + lq];
            pb.y = Vh[(size_t)(k + ko + 1) * HD + n0 + lq];
            o = wmma_f32(pa, pb, o);
        }
        if (wave >= 4) {
            #pragma unroll
            for (int r = 0; r < 8; ++r)
                outbuf[(wave & 3) * 256 + (rbase + r) * 16 + lq] = o[r];
        }
        __syncthreads();
        if (wave < 4) {
            float* op = Out + ((size_t)(h * SEQ) + q0) * HD;
            #pragma unroll
            for (int r = 0; r < 8; ++r) {
                float v = o[r] + outbuf[wave * 256 + (rbase + r) * 16 + lq];
                op[(size_t)(rbase + r) * HD + n0 + lq] = v;
            }
        }
    }
}

extern "C" void kernel_launch(void* const* d_in, const int* in_sizes, int n_in,
                              void* d_out, int out_size, void* d_ws, size_t ws_size,
                              hipStream_t stream) {
    const float* Q = (const float*)d_in[0];
    const float* K = (const float*)d_in[1];
    const float* V = (const float*)d_in[2];
    float* out     = (float*)d_out;
    float* student = out + (size_t)NH * SEQ * HD;
    float* teacher = student + (size_t)NH * SEQ * SEQ;

    (void)hipFuncSetAttribute((const void*)entmax_attn_kernel,
                              hipFuncAttributeMaxDynamicSharedMemorySize,
                              LDS_BYTES);

    hipLaunchKernelGGL(entmax_attn_kernel, dim3(NH * 128), dim3(256), LDS_BYTES,
                       stream, Q, K, V, out, student, teacher);
}